// PointNet_Plus_54494545052196
// MI455X (gfx1250) — compile-verified
//
#include <hip/hip_runtime.h>

typedef __attribute__((ext_vector_type(2))) float v2f;
typedef __attribute__((ext_vector_type(8))) float v8f;

#define B_ 8
#define N_ 2048
#define BN_ (B_ * N_)
#define R2_ (0.45f * 0.45f)

// ---------------------------------------------------------------------------
// Kernel 1: per-point features (wave per point, lane = channel slice)
//   H[pt, 0:64]   = relu(Wc1 . p + bc1)          (pointwise conv branch)
//   Fa[pt, 0:64]  = relu(W2a . relu(W1a.p+b1a) + b2a)
//   Fb[pt, 0:128] = relu(W2b . relu(W1b.p+b1b) + b2b)
// ---------------------------------------------------------------------------
__global__ void k_point_features(const float* __restrict__ x,
                                 const float* __restrict__ W1a, const float* __restrict__ b1a,
                                 const float* __restrict__ W2a, const float* __restrict__ b2a,
                                 const float* __restrict__ W1b, const float* __restrict__ b1b,
                                 const float* __restrict__ W2b, const float* __restrict__ b2b,
                                 const float* __restrict__ Wc1, const float* __restrict__ bc1,
                                 float* __restrict__ H,
                                 float* __restrict__ Fa,
                                 float* __restrict__ Fb) {
    const int wid  = (blockIdx.x * blockDim.x + threadIdx.x) >> 5;   // point id
    const int lane = threadIdx.x & 31;
    if (wid >= BN_) return;

    const float px = x[3 * wid + 0];
    const float py = x[3 * wid + 1];
    const float pz = x[3 * wid + 2];

    // ---- branch a: 3 -> 32 -> 64 ----
    float ha = fmaxf(W1a[lane * 3 + 0] * px + W1a[lane * 3 + 1] * py +
                     W1a[lane * 3 + 2] * pz + b1a[lane], 0.0f);
    float a0 = b2a[lane];
    float a1 = b2a[lane + 32];
    #pragma unroll
    for (int k = 0; k < 32; ++k) {
        float hk = __shfl(ha, k, 32);
        a0 += W2a[lane * 32 + k] * hk;
        a1 += W2a[(lane + 32) * 32 + k] * hk;
    }
    Fa[(size_t)wid * 64 + lane]      = fmaxf(a0, 0.0f);
    Fa[(size_t)wid * 64 + lane + 32] = fmaxf(a1, 0.0f);

    // ---- branch b: 3 -> 64 -> 128 ----
    float hb0 = fmaxf(W1b[lane * 3 + 0] * px + W1b[lane * 3 + 1] * py +
                      W1b[lane * 3 + 2] * pz + b1b[lane], 0.0f);
    float hb1 = fmaxf(W1b[(lane + 32) * 3 + 0] * px + W1b[(lane + 32) * 3 + 1] * py +
                      W1b[(lane + 32) * 3 + 2] * pz + b1b[lane + 32], 0.0f);
    float o0 = b2b[lane];
    float o1 = b2b[lane + 32];
    float o2 = b2b[lane + 64];
    float o3 = b2b[lane + 96];
    #pragma unroll
    for (int k = 0; k < 32; ++k) {
        float hk0 = __shfl(hb0, k, 32);      // hidden channel k
        float hk1 = __shfl(hb1, k, 32);      // hidden channel k+32
        o0 += W2b[(lane +  0) * 64 + k] * hk0 + W2b[(lane +  0) * 64 + 32 + k] * hk1;
        o1 += W2b[(lane + 32) * 64 + k] * hk0 + W2b[(lane + 32) * 64 + 32 + k] * hk1;
        o2 += W2b[(lane + 64) * 64 + k] * hk0 + W2b[(lane + 64) * 64 + 32 + k] * hk1;
        o3 += W2b[(lane + 96) * 64 + k] * hk0 + W2b[(lane + 96) * 64 + 32 + k] * hk1;
    }
    Fb[(size_t)wid * 128 + lane +  0] = fmaxf(o0, 0.0f);
    Fb[(size_t)wid * 128 + lane + 32] = fmaxf(o1, 0.0f);
    Fb[(size_t)wid * 128 + lane + 64] = fmaxf(o2, 0.0f);
    Fb[(size_t)wid * 128 + lane + 96] = fmaxf(o3, 0.0f);

    // ---- pointwise conv branch: channels 0..63 of H ----
    float c0 = fmaxf(Wc1[lane * 3 + 0] * px + Wc1[lane * 3 + 1] * py +
                     Wc1[lane * 3 + 2] * pz + bc1[lane], 0.0f);
    float c1 = fmaxf(Wc1[(lane + 32) * 3 + 0] * px + Wc1[(lane + 32) * 3 + 1] * py +
                     Wc1[(lane + 32) * 3 + 2] * pz + bc1[lane + 32], 0.0f);
    H[(size_t)wid * 128 + lane]      = c0;
    H[(size_t)wid * 128 + lane + 32] = c1;
}

// ---------------------------------------------------------------------------
// Kernel 2: ball-query + max-pool (wave per point)
//   Scan candidates j = 0..N-1 in index order; hit rank < 64 updates g1,
//   rank < 128 updates g2; early-exit at 128 hits.
//   Lane l owns g1 channels [2l,2l+1] and g2 channels [4l..4l+3].
//   Writes g1 -> H[pt, 64:128]; g2 -> out channels 256..383 (B,384,N layout).
// ---------------------------------------------------------------------------
__global__ void k_ball_maxpool(const float* __restrict__ x,
                               const float* __restrict__ Fa,
                               const float* __restrict__ Fb,
                               float* __restrict__ H,
                               float* __restrict__ out) {
    const int wid  = (blockIdx.x * blockDim.x + threadIdx.x) >> 5;   // global point
    const int lane = threadIdx.x & 31;
    if (wid >= BN_) return;

    const int b = wid / N_;
    const int i = wid - b * N_;
    const float* __restrict__ xb = x + (size_t)b * N_ * 3;

    const float cx = xb[i * 3 + 0];
    const float cy = xb[i * 3 + 1];
    const float cz = xb[i * 3 + 2];

    float g1x = 0.0f, g1y = 0.0f;                                // relu >= 0, >=1 neighbor
    float g2x = 0.0f, g2y = 0.0f, g2z = 0.0f, g2w = 0.0f;
    int cnt = 0;                                                 // uniform hit counter

    const float* __restrict__ FaB = Fa + (size_t)b * N_ * 64;
    const float* __restrict__ FbB = Fb + (size_t)b * N_ * 128;

    for (int c = 0; c < N_; c += 32) {
        const int j  = c + lane;
        const float dx = xb[j * 3 + 0] - cx;
        const float dy = xb[j * 3 + 1] - cy;
        const float dz = xb[j * 3 + 2] - cz;
        const bool hit = (dx * dx + dy * dy + dz * dz) <= R2_;
        unsigned m = __builtin_amdgcn_ballot_w32(hit);           // uniform (SGPR)

        while (m) {                                              // walk hits in index order
            const int p  = __builtin_ctz(m);
            m &= m - 1u;
            const int jj = c + p;
            const float4 fb = *reinterpret_cast<const float4*>(FbB + (size_t)jj * 128 + 4 * lane);
            g2x = fmaxf(g2x, fb.x);
            g2y = fmaxf(g2y, fb.y);
            g2z = fmaxf(g2z, fb.z);
            g2w = fmaxf(g2w, fb.w);
            if (cnt < 64) {
                const float2 fa = *reinterpret_cast<const float2*>(FaB + (size_t)jj * 64 + 2 * lane);
                g1x = fmaxf(g1x, fa.x);
                g1y = fmaxf(g1y, fa.y);
            }
            if (++cnt >= 128) break;
        }
        if (cnt >= 128) break;
    }

    // g1 -> H channels 64..127 (feeds the WMMA GEMM)
    float2* h2 = reinterpret_cast<float2*>(H + (size_t)wid * 128 + 64 + 2 * lane);
    *h2 = make_float2(g1x, g1y);

    // g2 -> output channels 256..383 at (b, ch, i)
    float* ob = out + ((size_t)b * 384 + 256) * N_ + i;
    ob[(size_t)(4 * lane + 0) * N_] = g2x;
    ob[(size_t)(4 * lane + 1) * N_] = g2y;
    ob[(size_t)(4 * lane + 2) * N_] = g2z;
    ob[(size_t)(4 * lane + 3) * N_] = g2w;
}

// ---------------------------------------------------------------------------
// Kernel 3: xf = relu(Wc2 @ H + bc2) via V_WMMA_F32_16X16X4_F32
//   Block = 512 threads = 16 waves; wave w owns output-channel tile m0=16w,
//   block owns 16 consecutive points. K-loop: 128 in steps of 4.
//   A (16x4 f32) = Wc2 tile, B (4x16 f32) = H^T tile:
//     lanes 0-15  hold (row, K=k0..k0+1), lanes 16-31 hold (row, K=k0+2..k0+3).
//   C/D layout: VGPR r -> channel m0 + r (lanes 0-15) / m0 + r + 8 (lanes 16-31),
//   lane%16 -> point. EXEC all ones (exact grid, no early return).
// ---------------------------------------------------------------------------
__global__ void k_gemm_wc2(const float* __restrict__ H,
                           const float* __restrict__ Wc2,
                           const float* __restrict__ bc2,
                           float* __restrict__ out) {
    const int lane = threadIdx.x & 31;
    const int w    = threadIdx.x >> 5;       // 0..15 -> channel tile
    const int m0   = w * 16;
    const int pt0  = blockIdx.x * 16;        // 16 consecutive points (never crosses batch)
    const int row  = lane & 15;
    const int hi   = lane >> 4;              // 0: K=k0,k0+1  1: K=k0+2,k0+3

    const float* __restrict__ Arow = Wc2 + (size_t)(m0 + row) * 128 + 2 * hi;
    const float* __restrict__ Brow = H   + (size_t)(pt0 + row) * 128 + 2 * hi;

    v8f acc = {};
    #pragma unroll
    for (int k0 = 0; k0 < 128; k0 += 4) {
        v2f a = *reinterpret_cast<const v2f*>(Arow + k0);
        v2f bb = *reinterpret_cast<const v2f*>(Brow + k0);
        acc = __builtin_amdgcn_wmma_f32_16x16x4_f32(
            /*neg_a=*/false, a, /*neg_b=*/false, bb,
            /*c_mod=*/(short)0, acc, /*reuse_a=*/false, /*reuse_b=*/false);
    }

    const int b  = pt0 / N_;                 // uniform within block
    const int i0 = pt0 - b * N_;
    #pragma unroll
    for (int r = 0; r < 8; ++r) {
        const int ch = m0 + r + 8 * hi;
        const float v = fmaxf(acc[r] + bc2[ch], 0.0f);
        out[((size_t)b * 384 + ch) * N_ + i0 + row] = v;
    }
}

// ---------------------------------------------------------------------------
// Launch
// ---------------------------------------------------------------------------
extern "C" void kernel_launch(void* const* d_in, const int* in_sizes, int n_in,
                              void* d_out, int out_size, void* d_ws, size_t ws_size,
                              hipStream_t stream) {
    const float* x   = (const float*)d_in[0];
    const float* W1a = (const float*)d_in[1];
    const float* b1a = (const float*)d_in[2];
    const float* W2a = (const float*)d_in[3];
    const float* b2a = (const float*)d_in[4];
    const float* W1b = (const float*)d_in[5];
    const float* b1b = (const float*)d_in[6];
    const float* W2b = (const float*)d_in[7];
    const float* b2b = (const float*)d_in[8];
    const float* Wc1 = (const float*)d_in[9];
    const float* bc1 = (const float*)d_in[10];
    const float* Wc2 = (const float*)d_in[11];
    const float* bc2 = (const float*)d_in[12];
    float* out = (float*)d_out;

    // workspace: H (BN x 128) | Fa (BN x 64) | Fb (BN x 128)  = 20 MB
    float* ws = (float*)d_ws;
    float* H  = ws;
    float* Fa = ws + (size_t)BN_ * 128;
    float* Fb = ws + (size_t)BN_ * 192;

    // 1) per-point features: BN waves, 8 waves/block
    k_point_features<<<BN_ / 8, 256, 0, stream>>>(x, W1a, b1a, W2a, b2a,
                                                  W1b, b1b, W2b, b2b,
                                                  Wc1, bc1, H, Fa, Fb);
    // 2) ball query + max-pool: BN waves, 8 waves/block
    k_ball_maxpool<<<BN_ / 8, 256, 0, stream>>>(x, Fa, Fb, H, out);
    // 3) WMMA f32 GEMM for Wc2: one block per 16 points, 16 waves/block
    k_gemm_wc2<<<BN_ / 16, 512, 0, stream>>>(H, Wc2, bc2, out);
}